// HybridNetworkGlobal_12292196401954
// MI455X (gfx1250) — compile-verified
//
#include <hip/hip_runtime.h>
#include <hip/hip_bf16.h>

// ---------------------------------------------------------------------------
// HybridNetworkGlobal forward for MI455X (gfx1250, wave32).
// Dense GEMMs: V_WMMA_F32_16X16X4_F32, compile-time shapes, weight fragments
// resident in VGPRs across 4 row tiles per wave, concat fused into A reads,
// bias folded into the WMMA accumulator init.
// Edge aggregation: fp32 global atomics (all hot data L2-resident).
// ---------------------------------------------------------------------------

#define NODE_COUNT 1024
#define N_GRAPHS   128
#define NN         (NODE_COUNT * N_GRAPHS)   // 131072 nodes
#define HID        16

typedef __attribute__((ext_vector_type(2))) float v2f;
typedef __attribute__((ext_vector_type(8))) float v8f;

// ---- order-preserving float<->uint key for atomic max over signed floats ----
__device__ __forceinline__ unsigned fkey(float f) {
  unsigned u = __float_as_uint(f);
  return (u & 0x80000000u) ? ~u : (u | 0x80000000u);
}
__device__ __forceinline__ float keyf(unsigned k) {
  unsigned u = (k & 0x80000000u) ? (k & 0x7fffffffu) : ~k;
  return __uint_as_float(u);
}

// ---------------------------------------------------------------------------
// A-element accessor for the fused-concat GEMM:
//   col c < DA           -> A1[row, c]
//   DA <= c < DA+DB      -> A2[row, c-DA]      (BC2: row -> row/NODE_COUNT)
//   c >= DA+DB (K pad)   -> 0
// ---------------------------------------------------------------------------
template <int DA, int DB, bool BC2>
__device__ __forceinline__ float aload(const float* __restrict__ A1,
                                       const float* __restrict__ A2,
                                       int row, int c) {
  constexpr int K = DA + DB;
  if (c >= K) return 0.0f;
  if constexpr (DB == 0) {
    return A1[(long)row * DA + c];
  } else {
    if (c < DA) return A1[(long)row * DA + c];
    const int rr = BC2 ? (row >> 10) : row;
    return A2[(long)rr * DB + (c - DA)];
  }
}

// ---------------------------------------------------------------------------
// C[nrows x NCOLS] = act([A1|A2][nrows x K] @ B[K x NCOLS] + bias)
// One wave : TILES consecutive 16-row tiles; B fragments live in VGPRs;
// bias pre-loaded into the accumulator (per-column, exact).
// WMMA f32 16x16x4 fragment layouts per CDNA5 ISA:
//   A (16x4): lanes 0-15 K={0,1}, lanes 16-31 K={2,3}
//   C/D (16x16): VGPR i -> row i (lanes 0-15) / row 8+i (lanes 16-31)
// ---------------------------------------------------------------------------
template <int DA, int DB, int NCOLS, bool RELU, bool BIAS, bool BC2, int TILES>
__global__ __launch_bounds__(256) void gemm_fused_kernel(
    const float* __restrict__ A1, const float* __restrict__ A2,
    const float* __restrict__ B, const float* __restrict__ bias,
    float* __restrict__ C, int nrows) {
  constexpr int K = DA + DB;
  constexpr int KF = (K + 3) / 4;   // K fragments
  constexpr int CT = NCOLS / 16;    // column tiles
  const int lane = threadIdx.x & 31;
  const int wave = threadIdx.x >> 5;
  const int half = lane >> 4;
  const int l = lane & 15;
  const int job = blockIdx.x * (blockDim.x >> 5) + wave;

  // Weight fragments: loaded once, resident for all row tiles of this wave.
  v2f bf[CT][KF];
#pragma unroll
  for (int c = 0; c < CT; ++c)
#pragma unroll
    for (int kk = 0; kk < KF; ++kk) {
      const int ka = kk * 4 + 2 * half;
      bf[c][kk].x = (ka     < K) ? B[ka * NCOLS + c * 16 + l] : 0.0f;
      bf[c][kk].y = (ka + 1 < K) ? B[(ka + 1) * NCOLS + c * 16 + l] : 0.0f;
    }
  float bb[CT];
#pragma unroll
  for (int c = 0; c < CT; ++c) bb[c] = BIAS ? bias[c * 16 + l] : 0.0f;

  for (int t = 0; t < TILES; ++t) {
    const int r0 = (job * TILES + t) * 16;
    if (r0 >= nrows) return;          // wave-uniform: EXEC stays all-ones
    const int row = r0 + l;
    v2f af[KF];
#pragma unroll
    for (int kk = 0; kk < KF; ++kk) {
      const int ka = kk * 4 + 2 * half;
      af[kk].x = aload<DA, DB, BC2>(A1, A2, row, ka);
      af[kk].y = aload<DA, DB, BC2>(A1, A2, row, ka + 1);
    }
#pragma unroll
    for (int c = 0; c < CT; ++c) {
      v8f acc;
      if constexpr (BIAS) {
#pragma unroll
        for (int i = 0; i < 8; ++i) acc[i] = bb[c];   // bias folded into C
      } else {
        acc = (v8f){};
      }
#pragma unroll
      for (int kk = 0; kk < KF; ++kk)
        acc = __builtin_amdgcn_wmma_f32_16x16x4_f32(
            /*neg_a=*/false, af[kk], /*neg_b=*/false, bf[c][kk],
            /*c_mod=*/(short)0, acc, /*reuse_a=*/false, /*reuse_b=*/false);
#pragma unroll
      for (int i = 0; i < 8; ++i) {
        float v = acc[i];
        if (RELU) v = fmaxf(v, 0.0f);
        C[(long)(r0 + i + 8 * half) * NCOLS + c * 16 + l] = v;
      }
    }
  }
}

// ---------------------------------------------------------------------------
// Concatenate [A | B] -> Cc (only used once, for the final x_comb)
// ---------------------------------------------------------------------------
__global__ void concat_kernel(const float* __restrict__ A, const float* __restrict__ B,
                              float* __restrict__ Cc, int n, int dA, int dB) {
  const int d = dA + dB;
  long i = (long)blockIdx.x * blockDim.x + threadIdx.x;
  if (i >= (long)n * d) return;
  int r = (int)(i / d), c = (int)(i % d);
  Cc[i] = (c < dA) ? A[(long)r * dA + c] : B[(long)r * dB + (c - dA)];
}

// ---------------------------------------------------------------------------
// Per-node: a_src/a_dst = h . att_{src,dst}; init softmax max-key, sum, outacc
// ---------------------------------------------------------------------------
__global__ void att_init_kernel(const float* __restrict__ h,
                                const float* __restrict__ att_src,
                                const float* __restrict__ att_dst,
                                float* __restrict__ a_src, float* __restrict__ a_dst,
                                unsigned* __restrict__ mkey, float* __restrict__ ssum,
                                float* __restrict__ outacc, int n) {
  int i = blockIdx.x * blockDim.x + threadIdx.x;
  if (i >= n) return;
  const float* hr = h + (long)i * HID;
  float as = 0.0f, ad = 0.0f;
#pragma unroll
  for (int j = 0; j < HID; ++j) {
    float v = hr[j];
    as += v * att_src[j];
    ad += v * att_dst[j];
  }
  a_src[i] = as;
  a_dst[i] = ad;
  mkey[i] = fkey(-3.0e38f);
  ssum[i] = 0.0f;
  float4* o = (float4*)(outacc + (long)i * HID);
  float4 z = make_float4(0.f, 0.f, 0.f, 0.f);
  o[0] = z; o[1] = z; o[2] = z; o[3] = z;
}

__device__ __forceinline__ void edge_sd(const int* __restrict__ ei, int e, int E,
                                        int& s, int& d) {
  if (e < E) { s = ei[e];     d = ei[E + e]; }
  else       { s = ei[e];     d = ei[e - E]; }   // flipped copy: ei[E+(e-E)] == ei[e]
}

__global__ void edge_max_kernel(const int* __restrict__ ei,
                                const float* __restrict__ a_src,
                                const float* __restrict__ a_dst,
                                unsigned* __restrict__ mkey, int E) {
  int e = blockIdx.x * blockDim.x + threadIdx.x;
  if (e >= 2 * E) return;
  int s, d; edge_sd(ei, e, E, s, d);
  float x = a_src[s] + a_dst[d];
  x = (x > 0.0f) ? x : 0.2f * x;                 // leaky_relu(0.2)
  atomicMax(&mkey[d], fkey(x));
}

__global__ void edge_exp_kernel(const int* __restrict__ ei,
                                const float* __restrict__ a_src,
                                const float* __restrict__ a_dst,
                                const unsigned* __restrict__ mkey,
                                float* __restrict__ ssum,
                                float* __restrict__ exbuf, int E) {
  int e = blockIdx.x * blockDim.x + threadIdx.x;
  if (e >= 2 * E) return;
  int s, d; edge_sd(ei, e, E, s, d);
  float x = a_src[s] + a_dst[d];
  x = (x > 0.0f) ? x : 0.2f * x;
  float ex = __expf(x - keyf(mkey[d]));
  exbuf[e] = ex;
  atomicAdd(&ssum[d], ex);
}

__global__ void edge_scatter_kernel(const int* __restrict__ ei,
                                    const float* __restrict__ h,
                                    const float* __restrict__ ssum,
                                    const float* __restrict__ exbuf,
                                    float* __restrict__ outacc, int E) {
  int e = blockIdx.x * blockDim.x + threadIdx.x;
  if (e >= 2 * E) return;
  int s, d; edge_sd(ei, e, E, s, d);
  float alpha = exbuf[e] / (ssum[d] + 1e-16f);
  const float4* hs = (const float4*)(h + (long)s * HID);
  float* od = outacc + (long)d * HID;
  float4 h0 = hs[0], h1 = hs[1], h2 = hs[2], h3 = hs[3];
  atomicAdd(od + 0,  h0.x * alpha); atomicAdd(od + 1,  h0.y * alpha);
  atomicAdd(od + 2,  h0.z * alpha); atomicAdd(od + 3,  h0.w * alpha);
  atomicAdd(od + 4,  h1.x * alpha); atomicAdd(od + 5,  h1.y * alpha);
  atomicAdd(od + 6,  h1.z * alpha); atomicAdd(od + 7,  h1.w * alpha);
  atomicAdd(od + 8,  h2.x * alpha); atomicAdd(od + 9,  h2.y * alpha);
  atomicAdd(od + 10, h2.z * alpha); atomicAdd(od + 11, h2.w * alpha);
  atomicAdd(od + 12, h3.x * alpha); atomicAdd(od + 13, h3.y * alpha);
  atomicAdd(od + 14, h3.z * alpha); atomicAdd(od + 15, h3.w * alpha);
}

// x_conv = relu(outacc + gat_b)
__global__ void conv_finish_kernel(const float* __restrict__ outacc,
                                   const float* __restrict__ gat_b,
                                   float* __restrict__ xconv, int n) {
  long i = (long)blockIdx.x * blockDim.x + threadIdx.x;
  if (i >= (long)n * HID) return;
  xconv[i] = fmaxf(outacc[i] + gat_b[i & (HID - 1)], 0.0f);
}

// w[i] = x_comb[i,:32] . wa_W + wa_b
__global__ void w_kernel(const float* __restrict__ xcomb, const float* __restrict__ waW,
                         const float* __restrict__ wab, float* __restrict__ w, int n) {
  int i = blockIdx.x * blockDim.x + threadIdx.x;
  if (i >= n) return;
  const float* xr = xcomb + (long)i * 32;
  float v = wab[0];
#pragma unroll
  for (int k = 0; k < 32; ++k) v += xr[k] * waW[k];
  w[i] = v;
}

// block per graph: softmax over 1024 nodes, weighted sum -> graph_emb[32]
__global__ void graph_emb_kernel(const float* __restrict__ xcomb,
                                 const float* __restrict__ w,
                                 float* __restrict__ emb) {
  __shared__ float red[256];
  __shared__ float semb[32];
  const int g = blockIdx.x, t = threadIdx.x;
  const float* wg = w + (long)g * NODE_COUNT;
  float m = -3.0e38f;
  for (int n = t; n < NODE_COUNT; n += 256) m = fmaxf(m, wg[n]);
  red[t] = m; __syncthreads();
  for (int off = 128; off > 0; off >>= 1) {
    if (t < off) red[t] = fmaxf(red[t], red[t + off]);
    __syncthreads();
  }
  m = red[0]; __syncthreads();
  float s = 0.0f;
  for (int n = t; n < NODE_COUNT; n += 256) s += __expf(wg[n] - m);
  red[t] = s; __syncthreads();
  for (int off = 128; off > 0; off >>= 1) {
    if (t < off) red[t] += red[t + off];
    __syncthreads();
  }
  s = red[0];
  if (t < 32) semb[t] = 0.0f;
  __syncthreads();
  const float inv = 1.0f / s;
  float acc[32];
#pragma unroll
  for (int k = 0; k < 32; ++k) acc[k] = 0.0f;
  for (int n = t; n < NODE_COUNT; n += 256) {
    float p = __expf(wg[n] - m) * inv;
    const float* xr = xcomb + ((long)g * NODE_COUNT + n) * 32;
#pragma unroll
    for (int k = 0; k < 32; ++k) acc[k] += p * xr[k];
  }
#pragma unroll
  for (int k = 0; k < 32; ++k) atomicAdd(&semb[k], acc[k]);
  __syncthreads();
  if (t < 32) emb[(long)g * 32 + t] = semb[t];
}

// X[i] = h1[i,:32] . a2_W + a2_b
__global__ void a2_kernel(const float* __restrict__ h1, const float* __restrict__ a2W,
                          const float* __restrict__ a2b, float* __restrict__ out, int n) {
  int i = blockIdx.x * blockDim.x + threadIdx.x;
  if (i >= n) return;
  const float* hr = h1 + (long)i * 32;
  float v = a2b[0];
#pragma unroll
  for (int k = 0; k < 32; ++k) v += hr[k] * a2W[k];
  out[i] = v;
}

// V[g] = relu(emb[g] @ v1_W + v1_b) @ v2_W + v2_b
__global__ void v_head_kernel(const float* __restrict__ emb,
                              const float* __restrict__ v1W, const float* __restrict__ v1b,
                              const float* __restrict__ v2W, const float* __restrict__ v2b,
                              float* __restrict__ out, int ngraphs) {
  int g = blockIdx.x * blockDim.x + threadIdx.x;
  if (g >= ngraphs) return;
  const float* e = emb + (long)g * 32;
  float v = v2b[0];
#pragma unroll
  for (int j = 0; j < 16; ++j) {
    float t = v1b[j];
#pragma unroll
    for (int k = 0; k < 32; ++k) t += e[k] * v1W[k * 16 + j];
    v += fmaxf(t, 0.0f) * v2W[j];
  }
  out[g] = v;
}

// ---------------------------------------------------------------------------
// Host launch. Param pointers follow JAX pytree flattening (sorted dict keys):
//  0:x 1:edge_index 2:edge_attr 3:a1_W 4:a1_b 5:a2_W 6:a2_b
//  7+6*l .. : per layer l: att_dst, att_src, gat_W, gat_b, lin_W, lin_b
//  31:v1_W 32:v1_b 33:v2_W 34:v2_b 35:wa_W 36:wa_b
// ---------------------------------------------------------------------------
extern "C" void kernel_launch(void* const* d_in, const int* in_sizes, int n_in,
                              void* d_out, int out_size, void* d_ws, size_t ws_size,
                              hipStream_t stream) {
  const float* x  = (const float*)d_in[0];
  const int*   ei = (const int*)d_in[1];
  const int E = in_sizes[1] / 2;            // input edge count (doubled in kernels)
  const int E2 = 2 * E;

  const float* a1W = (const float*)d_in[3];
  const float* a1b = (const float*)d_in[4];
  const float* a2W = (const float*)d_in[5];
  const float* a2b = (const float*)d_in[6];
  const float *attD[4], *attS[4], *gatW[4], *gatB[4], *linW[4], *linB[4];
  for (int l = 0; l < 4; ++l) {
    int b = 7 + 6 * l;
    attD[l] = (const float*)d_in[b + 0];
    attS[l] = (const float*)d_in[b + 1];
    gatW[l] = (const float*)d_in[b + 2];
    gatB[l] = (const float*)d_in[b + 3];
    linW[l] = (const float*)d_in[b + 4];
    linB[l] = (const float*)d_in[b + 5];
  }
  const float* v1W = (const float*)d_in[31];
  const float* v1b = (const float*)d_in[32];
  const float* v2W = (const float*)d_in[33];
  const float* v2b = (const float*)d_in[34];
  const float* waW = (const float*)d_in[35];
  const float* wab = (const float*)d_in[36];

  // ---- workspace carve ----
  char* wp = (char*)d_ws;
  auto alloc = [&](size_t bytes) -> void* {
    void* p = (void*)wp;
    wp += (bytes + 255) & ~(size_t)255;
    return p;
  };
  float*    xlin   = (float*)alloc((size_t)NN * HID * 4);
  float*    xconv  = (float*)alloc((size_t)NN * HID * 4);
  float*    xcomb  = (float*)alloc((size_t)NN * 32 * 4);
  float*    hbuf   = (float*)alloc((size_t)NN * HID * 4);
  float*    asrc   = (float*)alloc((size_t)NN * 4);
  float*    adst   = (float*)alloc((size_t)NN * 4);
  unsigned* mkey   = (unsigned*)alloc((size_t)NN * 4);
  float*    ssum   = (float*)alloc((size_t)NN * 4);
  float*    outacc = (float*)alloc((size_t)NN * HID * 4);
  float*    exbuf  = (float*)alloc((size_t)E2 * 4);
  float*    wbuf   = (float*)alloc((size_t)NN * 4);
  float*    emb    = (float*)alloc((size_t)N_GRAPHS * 32 * 4);
  float*    h1     = (float*)alloc((size_t)NN * 32 * 4);

  const int TB = 256;
  const int nodeBlocks = (NN + TB - 1) / TB;
  const int edgeBlocks = (E2 + TB - 1) / TB;
  constexpr int TILES = 4;                       // 16-row tiles per wave
  const int tilesTotal = NN / 16;                // 8192
  const int gemmBlocks =
      ((tilesTotal + TILES - 1) / TILES + (TB / 32) - 1) / (TB / 32);  // 256

  // ---------------- layer 0 (d_in = 2) ----------------
  gemm_fused_kernel<2, 2, 16, false, false, false, TILES>
      <<<gemmBlocks, TB, 0, stream>>>(x, x, gatW[0], nullptr, hbuf, NN);
  att_init_kernel<<<nodeBlocks, TB, 0, stream>>>(
      hbuf, attS[0], attD[0], asrc, adst, mkey, ssum, outacc, NN);
  edge_max_kernel<<<edgeBlocks, TB, 0, stream>>>(ei, asrc, adst, mkey, E);
  edge_exp_kernel<<<edgeBlocks, TB, 0, stream>>>(ei, asrc, adst, mkey, ssum, exbuf, E);
  edge_scatter_kernel<<<edgeBlocks, TB, 0, stream>>>(ei, hbuf, ssum, exbuf, outacc, E);
  conv_finish_kernel<<<(NN * HID + TB - 1) / TB, TB, 0, stream>>>(outacc, gatB[0], xconv, NN);
  gemm_fused_kernel<2, 0, 16, true, true, false, TILES>
      <<<gemmBlocks, TB, 0, stream>>>(x, nullptr, linW[0], linB[0], xlin, NN);

  // ---------------- layers 1..3 (d_in = 16) ----------------
  for (int l = 1; l < 4; ++l) {
    gemm_fused_kernel<16, 16, 16, false, false, false, TILES>
        <<<gemmBlocks, TB, 0, stream>>>(xlin, xconv, gatW[l], nullptr, hbuf, NN);
    att_init_kernel<<<nodeBlocks, TB, 0, stream>>>(
        hbuf, attS[l], attD[l], asrc, adst, mkey, ssum, outacc, NN);
    edge_max_kernel<<<edgeBlocks, TB, 0, stream>>>(ei, asrc, adst, mkey, E);
    edge_exp_kernel<<<edgeBlocks, TB, 0, stream>>>(ei, asrc, adst, mkey, ssum, exbuf, E);
    edge_scatter_kernel<<<edgeBlocks, TB, 0, stream>>>(ei, hbuf, ssum, exbuf, outacc, E);
    conv_finish_kernel<<<(NN * HID + TB - 1) / TB, TB, 0, stream>>>(outacc, gatB[l], xconv, NN);
    // x_lin = relu(x_lin @ lin_W + lin_b); in-place is safe: each wave reads
    // only its own 16-row tile fully before writing it.
    gemm_fused_kernel<16, 0, 16, true, true, false, TILES>
        <<<gemmBlocks, TB, 0, stream>>>(xlin, nullptr, linW[l], linB[l], xlin, NN);
  }

  // ---------------- readout ----------------
  concat_kernel<<<(int)(((long)NN * 32 + TB - 1) / TB), TB, 0, stream>>>(
      xlin, xconv, xcomb, NN, HID, HID);
  w_kernel<<<nodeBlocks, TB, 0, stream>>>(xcomb, waW, wab, wbuf, NN);
  graph_emb_kernel<<<N_GRAPHS, TB, 0, stream>>>(xcomb, wbuf, emb);
  // h1 = relu([x_comb | emb(bcast)] @ a1_W + a1_b)  — concat fused via BC2
  gemm_fused_kernel<32, 32, 32, true, true, true, TILES>
      <<<gemmBlocks, TB, 0, stream>>>(xcomb, emb, a1W, a1b, h1, NN);
  // X head -> d_out[0..N)
  a2_kernel<<<nodeBlocks, TB, 0, stream>>>(h1, a2W, a2b, (float*)d_out, NN);
  // V head -> d_out[N..N+128)
  v_head_kernel<<<1, 128, 0, stream>>>(emb, v1W, v1b, v2W, v2b,
                                       (float*)d_out + NN, N_GRAPHS);
}